// NewSampler_63170378989663
// MI455X (gfx1250) — compile-verified
//
#include <hip/hip_runtime.h>
#include <cstdint>
#include <cstddef>

// NewSampler fine-sampling kernel for gfx1250 (MI455X).
// One wave32 per ray; 8 waves (8 rays) per 256-thread block.
// Bandwidth-bound (~3.1 KB traffic/ray) -> no WMMA; uses the CDNA5
// async-tensor paths both ways:
//   inputs : global_load_async_to_lds_b32  + s_wait_asynccnt
//   z/s out: global_store_async_from_lds_b128 (LDS -> L2, no VGPR trip)

#define WAVES_PER_BLOCK 8
#define SLAB 384           // floats of LDS per wave
// slab layout (float offsets): sv[0..63], wt[64..127], mid[128..190],
//                              cdf[192..254], buf[256..383]

__device__ __forceinline__ unsigned rotl32(unsigned x, unsigned r) {
  return (x << r) | (x >> (32u - r));
}

// Exact JAX threefry2x32 for key = jax.random.key(42) -> (k0=0, k1=42).
// Flat sample index g in [0, n); counters are (g, g+n/2) for g < n/2
// (take x0), else (g-n/2, g) (take x1). Returns uniform in [0,1).
__device__ __forceinline__ float threefry_uniform(unsigned g, unsigned nhalf) {
  unsigned c0, c1;
  bool hi;
  if (g < nhalf) { c0 = g;         c1 = g + nhalf; hi = false; }
  else           { c0 = g - nhalf; c1 = g;         hi = true;  }
  const unsigned k0 = 0u, k1 = 42u;
  const unsigned k2 = 0x1BD11BDAu ^ k0 ^ k1;
  unsigned x0 = c0 + k0, x1 = c1 + k1;
#define TF_R4(a, b, c, d)                      \
  x0 += x1; x1 = rotl32(x1, a); x1 ^= x0;      \
  x0 += x1; x1 = rotl32(x1, b); x1 ^= x0;      \
  x0 += x1; x1 = rotl32(x1, c); x1 ^= x0;      \
  x0 += x1; x1 = rotl32(x1, d); x1 ^= x0;
  TF_R4(13, 15, 26, 6)   x0 += k1; x1 += k2 + 1u;
  TF_R4(17, 29, 16, 24)  x0 += k2; x1 += k0 + 2u;
  TF_R4(13, 15, 26, 6)   x0 += k0; x1 += k1 + 3u;
  TF_R4(17, 29, 16, 24)  x0 += k1; x1 += k2 + 4u;
  TF_R4(13, 15, 26, 6)   x0 += k2; x1 += k0 + 5u;
#undef TF_R4
  unsigned bits = hi ? x1 : x0;
  return __uint_as_float((bits >> 9) | 0x3f800000u) - 1.0f;
}

__global__ __launch_bounds__(256) void newsampler_fine_kernel(
    const float* __restrict__ rays_o,   // (B, 3)
    const float* __restrict__ rays_d,   // (B, 3)
    const float* __restrict__ s_vals,   // (B, 64) sorted
    const float* __restrict__ weights,  // (B, 64)
    float* __restrict__ out_pts,        // (B, 128, 3)
    float* __restrict__ out_z,          // (B, 128)
    float* __restrict__ out_s,          // (B, 128)
    int nrays) {
  __shared__ __align__(16) float smem[WAVES_PER_BLOCK * SLAB];

  const int lane = threadIdx.x & 31;
  const int wid  = threadIdx.x >> 5;
  const int ray  = blockIdx.x * WAVES_PER_BLOCK + wid;
  if (ray >= nrays) return;

  float* sv  = &smem[wid * SLAB + 0];    // 64 coarse depths
  float* wt  = &smem[wid * SLAB + 64];   // 64 raw weights
  float* mid = &smem[wid * SLAB + 128];  // 63 bin edges
  float* cdf = &smem[wid * SLAB + 192];  // 63 cdf entries (cdf[0]=0)
  float* buf = &smem[wid * SLAB + 256];  // 128-wide sort buffer

  // ---- Stage s_vals/weights into LDS via CDNA5 async-tensor path ----
  // GVS addressing: SGPR 64-bit base + per-lane VGPR byte offset.
  // smem is the only __shared__ object -> its LDS base offset is 0,
  // so LDS byte addresses are formed arithmetically.
  {
    unsigned goff0 = (unsigned)((ray * 64 + lane) * 4);
    unsigned goff1 = goff0 + 32u * 4u;
    unsigned lds_sv0 = (unsigned)((wid * SLAB + 0  + lane) * 4);
    unsigned lds_sv1 = lds_sv0 + 128u;
    unsigned lds_wt0 = (unsigned)((wid * SLAB + 64 + lane) * 4);
    unsigned lds_wt1 = lds_wt0 + 128u;
    asm volatile(
        "global_load_async_to_lds_b32 %0, %4, %6\n\t"
        "global_load_async_to_lds_b32 %1, %5, %6\n\t"
        "global_load_async_to_lds_b32 %2, %4, %7\n\t"
        "global_load_async_to_lds_b32 %3, %5, %7\n\t"
        "s_wait_asynccnt 0"
        :
        : "v"(lds_sv0), "v"(lds_sv1), "v"(lds_wt0), "v"(lds_wt1),
          "v"(goff0), "v"(goff1), "s"(s_vals), "s"(weights)
        : "memory");
  }
  __builtin_amdgcn_wave_barrier();

  // ---- Bin midpoints: mid[k] = 0.5*(sv[k]+sv[k+1]), k = 0..62 ----
#pragma unroll
  for (int q = 0; q < 2; ++q) {
    int k = lane + q * 32;
    if (k < 63) mid[k] = 0.5f * (sv[k] + sv[k + 1]);
  }

  // ---- CDF over 62 interior weights via wave32 shuffle scan ----
  // lane i owns weight indices 2i, 2i+1 (w[k] = weights[k+1] + 1e-6).
  float a = 0.0f, b = 0.0f;
  int ka = 2 * lane, kb = 2 * lane + 1;
  if (ka < 62) a = wt[ka + 1] + 1e-6f;
  if (kb < 62) b = wt[kb + 1] + 1e-6f;
  float local = a + b;
  float incl = local;
#pragma unroll
  for (int d = 1; d < 32; d <<= 1) {
    float y = __shfl_up(incl, d, 32);
    if (lane >= d) incl += y;
  }
  float total = __shfl(incl, 31, 32);
  float inv = 1.0f / total;
  float excl = incl - local;
  if (lane == 0) cdf[0] = 0.0f;
  if (ka < 62) cdf[ka + 1] = (excl + a) * inv;
  if (kb < 62) cdf[kb + 1] = (excl + a + b) * inv;
  __builtin_amdgcn_wave_barrier();

  // ---- 64 fine samples: threefry uniform + inverse-CDF lerp ----
  const unsigned nhalf = (unsigned)nrays * 32u;  // (B*64)/2
#pragma unroll
  for (int q = 0; q < 2; ++q) {
    int j = lane + q * 32;
    unsigned g = (unsigned)ray * 64u + (unsigned)j;
    float u = threefry_uniform(g, nhalf);
    // searchsorted(cdf[0:62], u, side='right'); cdf[0]=0 <= u always.
    int lo = 0, hi = 62;
    while (lo < hi) {
      int m = (lo + hi) >> 1;
      bool le = (cdf[m] <= u);
      lo = le ? (m + 1) : lo;
      hi = le ? hi : m;
    }
    int above = lo;           // in [1, 62]
    int below = above - 1;
    float cb = cdf[below], ca = cdf[above];
    float bb = mid[below], ba = mid[above];
    float denom = ca - cb;
    denom = (denom < 1e-6f) ? 1.0f : denom;
    float t = (u - cb) / denom;
    buf[j] = bb + t * (ba - bb + 1e-6f);
  }
  // append the 64 coarse depths
  buf[64 + lane] = sv[lane];
  buf[96 + lane] = sv[lane + 32];
  __builtin_amdgcn_wave_barrier();

  // ---- 128-element bitonic sort in LDS (wave-lockstep, no barriers) ----
  for (int k = 2; k <= 128; k <<= 1) {
    for (int j = k >> 1; j >= 1; j >>= 1) {
#pragma unroll
      for (int q = 0; q < 4; ++q) {
        int i = lane + q * 32;
        if ((i & j) == 0) {
          int p = i ^ j;  // p = i + j < 128
          float x = buf[i];
          float y = buf[p];
          bool asc = ((i & k) == 0);
          bool sw = asc ? (x > y) : (x < y);
          if (sw) { buf[i] = y; buf[p] = x; }
        }
      }
      __builtin_amdgcn_wave_barrier();
    }
  }

  // ---- z and s outputs: direct LDS -> global via async store ----
  // Sorted depths already live in LDS; copy them out with
  // global_store_async_from_lds_b128 (no VGPR round-trip). s_wait_dscnt 0
  // first: ASYNCcnt ops are not ordered against the sort's DS writes.
  {
    unsigned lds_buf = (unsigned)((wid * SLAB + 256 + lane * 4) * 4);
    unsigned zoff = (unsigned)(ray * 512 + lane * 16);  // byte offset
    asm volatile(
        "s_wait_dscnt 0\n\t"
        "global_store_async_from_lds_b128 %0, %1, %2\n\t"
        "global_store_async_from_lds_b128 %0, %1, %3"
        :
        : "v"(zoff), "v"(lds_buf), "s"(out_z), "s"(out_s)
        : "memory");
    // s_endpgm's implicit wait-idle drains ASYNCcnt before completion.
  }

  // ---- pts: lane i owns samples 4i..4i+3 -> 48 contiguous bytes ----
  const int uray = __builtin_amdgcn_readfirstlane(ray);  // scalarize o/d loads
  float ox = rays_o[uray * 3 + 0], oy = rays_o[uray * 3 + 1], oz = rays_o[uray * 3 + 2];
  float dx = rays_d[uray * 3 + 0], dy = rays_d[uray * 3 + 1], dz = rays_d[uray * 3 + 2];

  const float4 zq = *(const float4*)&buf[lane * 4];  // ds_load_b128
  float4 p0 = make_float4(ox + dx * zq.x, oy + dy * zq.x, oz + dz * zq.x,
                          ox + dx * zq.y);
  float4 p1 = make_float4(oy + dy * zq.y, oz + dz * zq.y,
                          ox + dx * zq.z, oy + dy * zq.z);
  float4 p2 = make_float4(oz + dz * zq.z,
                          ox + dx * zq.w, oy + dy * zq.w, oz + dz * zq.w);
  float* pp = out_pts + (size_t)ray * 384 + lane * 12;  // 16B-aligned
  *(float4*)(pp + 0) = p0;
  *(float4*)(pp + 4) = p1;
  *(float4*)(pp + 8) = p2;
}

extern "C" void kernel_launch(void* const* d_in, const int* in_sizes, int n_in,
                              void* d_out, int out_size, void* d_ws, size_t ws_size,
                              hipStream_t stream) {
  (void)n_in; (void)out_size; (void)d_ws; (void)ws_size;
  const float* rays_o  = (const float*)d_in[0];
  const float* rays_d  = (const float*)d_in[1];
  const float* s_vals  = (const float*)d_in[2];
  const float* weights = (const float*)d_in[3];
  const int nrays = in_sizes[0] / 3;  // B = 131072

  float* out = (float*)d_out;
  float* out_pts = out;                                        // B*128*3
  float* out_z   = out + (size_t)nrays * 384;                  // B*128
  float* out_s   = out + (size_t)nrays * 384 + (size_t)nrays * 128;

  dim3 block(256);
  dim3 grid((nrays + WAVES_PER_BLOCK - 1) / WAVES_PER_BLOCK);
  newsampler_fine_kernel<<<grid, block, 0, stream>>>(
      rays_o, rays_d, s_vals, weights, out_pts, out_z, out_s, nrays);
}